// LearnableSqueezePositionalEncoding_64613488001373
// MI455X (gfx1250) — compile-verified
//
#include <hip/hip_runtime.h>
#include <hip/hip_bf16.h>

typedef __attribute__((ext_vector_type(2))) float v2f;
typedef __attribute__((ext_vector_type(4))) float v4f;
typedef __attribute__((ext_vector_type(8))) float v8f;

#define N_CH   96      // embedding channels per table
#define N_OUT  192     // output channels
#define N_ROWS 128     // rows per table (64 coords * SQ=2)

// ---------------------------------------------------------------------------
// Kernel 1: T[t] = W[t] (128x96) @ P[t] (96x192), bias folded into T[0].
// One wave (32 threads) per 16x16 output tile, fp32 WMMA 16x16x4.
// Tiles: 3 tables * (128/16) * (192/16) = 3 * 8 * 12 = 288 blocks.
// ---------------------------------------------------------------------------
__global__ __launch_bounds__(32)
void lspe_precompute_wmma(const float* __restrict__ W0,
                          const float* __restrict__ W1,
                          const float* __restrict__ W2,
                          const float* __restrict__ proj_w,   // (288,192)
                          const float* __restrict__ proj_b,   // (192,)
                          float* __restrict__ T) {            // 3*128*192
    const int bid   = blockIdx.x;
    const int t     = bid / 96;          // which table
    const int rem   = bid % 96;
    const int tileM = rem / 12;          // 0..7
    const int tileN = rem % 12;          // 0..11
    const int m0 = tileM * 16;
    const int n0 = tileN * 16;

    const float* W = (t == 0) ? W0 : (t == 1) ? W1 : W2;
    const float* P = proj_w + (size_t)t * N_CH * N_OUT;  // 96x192 row block

    const int lane   = threadIdx.x;      // 0..31
    const int lane15 = lane & 15;
    const int khalf  = (lane >> 4) * 2;  // 0 or 2

    // A fragment rows: M = m0 + lane15 ; K within step: khalf, khalf+1
    const float* Arow = W + (size_t)(m0 + lane15) * N_CH;
    // B fragment cols: N = n0 + lane15 ; K rows: khalf, khalf+1
    const float* Bcol = P + n0 + lane15;

    v8f acc = {};
#pragma unroll
    for (int k = 0; k < N_CH; k += 4) {
        v2f a, b;
        a.x = Arow[k + khalf + 0];
        a.y = Arow[k + khalf + 1];
        b.x = Bcol[(size_t)(k + khalf + 0) * N_OUT];
        b.y = Bcol[(size_t)(k + khalf + 1) * N_OUT];
        acc = __builtin_amdgcn_wmma_f32_16x16x4_f32(
            /*neg_a=*/false, a, /*neg_b=*/false, b,
            /*c_mod=*/(short)0, acc, /*reuse_a=*/false, /*reuse_b=*/false);
    }

    float bias = (t == 0) ? proj_b[n0 + lane15] : 0.0f;

    float* Tout = T + (size_t)t * N_ROWS * N_OUT;
    const int rowBase = m0 + (lane >> 4) * 8;   // C layout: VGPR v -> M = v (+8 hi half)
#pragma unroll
    for (int v = 0; v < 8; ++v) {
        Tout[(size_t)(rowBase + v) * N_OUT + n0 + lane15] = acc[v] + bias;
    }
}

// ---------------------------------------------------------------------------
// Kernel 2: streaming gather-add.
// out[(n*8+combo)*192 + c] = T0[x*2+i][c] + T1[y*2+j][c] + T2[z*2+k][c]
// (bias already folded into T0). block = (48 quads, 8 corners) = 384 threads,
// one block per voxel. 128-bit coalesced nontemporal stores.
// ---------------------------------------------------------------------------
__global__ __launch_bounds__(384)
void lspe_gather_add(const int* __restrict__ coords,   // (N,3)
                     const float* __restrict__ T,      // 3*128*192
                     float* __restrict__ out) {
    const int n     = blockIdx.x;
    const int q     = threadIdx.x;   // 0..47  (float4 index within row)
    const int combo = threadIdx.y;   // 0..7   (i,j,k bits)
    const int i = (combo >> 2) & 1;
    const int j = (combo >> 1) & 1;
    const int k =  combo       & 1;

    const int x = coords[n * 3 + 0];
    const int y = coords[n * 3 + 1];
    const int z = coords[n * 3 + 2];

    const v4f* T0 = (const v4f*)(T)                      + (size_t)(x * 2 + i) * (N_OUT / 4);
    const v4f* T1 = (const v4f*)(T + 1 * N_ROWS * N_OUT) + (size_t)(y * 2 + j) * (N_OUT / 4);
    const v4f* T2 = (const v4f*)(T + 2 * N_ROWS * N_OUT) + (size_t)(z * 2 + k) * (N_OUT / 4);

    const v4f a = T0[q];
    const v4f b = T1[q];
    const v4f c = T2[q];
    const v4f r = a + b + c;

    v4f* dst = (v4f*)out + (size_t)(n * 8 + combo) * (N_OUT / 4) + q;
    __builtin_nontemporal_store(r, dst);
}

// ---------------------------------------------------------------------------
extern "C" void kernel_launch(void* const* d_in, const int* in_sizes, int n_in,
                              void* d_out, int out_size, void* d_ws, size_t ws_size,
                              hipStream_t stream) {
    const int*   coords = (const int*)  d_in[0];   // (N,3) int32
    const float* W0     = (const float*)d_in[1];   // (128,96)
    const float* W1     = (const float*)d_in[2];   // (128,96)
    const float* W2     = (const float*)d_in[3];   // (128,96)
    const float* proj_w = (const float*)d_in[4];   // (288,192)
    const float* proj_b = (const float*)d_in[5];   // (192,)
    float*       out    = (float*)d_out;
    float*       T      = (float*)d_ws;            // 3*128*192 floats = 294912 B

    const int N = in_sizes[0] / 3;

    // Stage 1: tiny fp32 WMMA GEMMs -> fused tables (bias folded into T0).
    lspe_precompute_wmma<<<dim3(3 * 8 * 12), dim3(32), 0, stream>>>(
        W0, W1, W2, proj_w, proj_b, T);

    // Stage 2: HBM-bound gather/add stream (402 MB of stores).
    lspe_gather_add<<<dim3(N), dim3(48, 8), 0, stream>>>(coords, T, out);
}